// GSPNLayer_89618787598567
// MI455X (gfx1250) — compile-verified
//
#include <hip/hip_runtime.h>
#include <math.h>

// ---------------------------------------------------------------------------
// GSPN layer for MI455X (gfx1250): fp32 WMMA 16x16x4 for all 1x1 convs,
// coalesced sequential scans, folded merge GEMM. Bandwidth-bound workload
// (~0.7GB traffic vs ~15 GFLOP fp32) -> keep fp32, saturate HBM.
// ---------------------------------------------------------------------------

typedef float v2f __attribute__((ext_vector_type(2)));
typedef float v8f __attribute__((ext_vector_type(8)));

#define SDIM   256
#define CDIM   128
#define CRDIM  32
#define NP     64            // pixels per tile
#define PPLANE 65536         // SDIM*SDIM
#define BDIM   2

__device__ __forceinline__ v8f wmma_f32(v2f a, v2f b, v8f c) {
    return __builtin_amdgcn_wmma_f32_16x16x4_f32(false, a, false, b,
                                                 (short)0, c, false, false);
}

__device__ __forceinline__ float sigmoidf_(float x) {
    return 1.0f / (1.0f + expf(-x));
}

// ---------------------------------------------------------------------------
// Kernel 1: per 64-pixel tile, compute red = Wr*x (WMMA, K=128), then
// u = Wu*red, v = (Wl*red + bl) * x (WMMA, K=32), and the normalized
// sigmoid scan factors sB[b,i,j] (mask on i) and sAT[b,j,i] (mask on j).
// ---------------------------------------------------------------------------
__global__ __launch_bounds__(256) void k1_pointwise(
    const float* __restrict__ x,
    const float* __restrict__ w_reduce, const float* __restrict__ b_reduce,
    const float* __restrict__ w_u,      const float* __restrict__ b_u,
    const float* __restrict__ w_lam,    const float* __restrict__ b_lam,
    const float* __restrict__ w_w,      const float* __restrict__ b_w,
    float* __restrict__ u_out, float* __restrict__ v_out,
    float* __restrict__ sAT,   float* __restrict__ sB)
{
    __shared__ float xs[CDIM * NP];   // 32 KB: x tile, xs[c*NP + p]
    __shared__ float rs[CRDIM * NP];  //  8 KB: red tile, rs[r*NP + p]

    const int tid  = threadIdx.x;
    const int lane = tid & 31;
    const int wv   = tid >> 5;          // 8 waves
    const int hl   = lane >> 4;         // lane half (0/1)
    const int ln   = lane & 15;

    const int tile = blockIdx.x;        // 2048 tiles
    const int b    = tile >> 10;        // 1024 tiles per batch image
    const int p0   = (tile & 1023) * NP;
    const float* xb = x + (size_t)b * CDIM * PPLANE;

    // ---- stage x tile: 128 channels x 64 pixels -------------------------
    #pragma unroll
    for (int it = 0; it < 8; ++it) {
        int li = it * 256 + tid;               // 0..2047
        int c  = li >> 4;                      // 16 float4 per channel
        int f  = (li & 15) << 2;
        *(float4*)(xs + c * NP + f) =
            *(const float4*)(xb + (size_t)c * PPLANE + p0 + f);
    }
    __syncthreads();

    // ---- red GEMM: [32x128]x[128x64], 8 output tiles, 1 per wave --------
    {
        const int m0 = (wv >> 2) * 16;
        const int n0 = (wv & 3) * 16;
        v8f acc = {};
        for (int ks = 0; ks < 32; ++ks) {      // K = 128, 4 per WMMA
            int c4 = 4 * ks;
            int kb = c4 + 2 * hl;
            v2f a  = *(const v2f*)(w_reduce + (m0 + ln) * CDIM + kb);
            v2f bb;
            bb.x = xs[kb * NP + n0 + ln];
            bb.y = xs[(kb + 1) * NP + n0 + ln];
            acc = wmma_f32(a, bb, acc);
        }
        #pragma unroll
        for (int g = 0; g < 8; ++g) {
            int m = m0 + g + 8 * hl;
            rs[m * NP + n0 + ln] = acc[g] + b_reduce[m];
        }
    }
    __syncthreads();

    // ---- u / lam GEMMs: [128x32]x[32x64]; share B fragments -------------
    for (int q = 0; q < 4; ++q) {
        int t  = q * 8 + wv;                   // 0..31
        int m0 = (t >> 2) * 16;
        int n0 = (t & 3) * 16;
        v8f au = {};
        v8f al = {};
        for (int ks = 0; ks < 8; ++ks) {       // K = 32
            int c4 = 4 * ks;
            int kb = c4 + 2 * hl;
            v2f a_u = *(const v2f*)(w_u   + (m0 + ln) * CRDIM + kb);
            v2f a_l = *(const v2f*)(w_lam + (m0 + ln) * CRDIM + kb);
            v2f bb;
            bb.x = rs[kb * NP + n0 + ln];
            bb.y = rs[(kb + 1) * NP + n0 + ln];
            au = wmma_f32(a_u, bb, au);
            al = wmma_f32(a_l, bb, al);
        }
        int gp = p0 + n0 + ln;
        #pragma unroll
        for (int g = 0; g < 8; ++g) {
            int m = m0 + g + 8 * hl;
            size_t off = (size_t)(b * CDIM + m) * PPLANE + gp;
            u_out[off] = au[g] + b_u[m];
            v_out[off] = (al[g] + b_lam[m]) * xs[m * NP + n0 + ln];
        }
    }

    // ---- pre -> sigmoid -> normalized scan factors ----------------------
    if (tid < NP) {
        int p = tid;
        float pr0 = b_w[0], pr1 = b_w[1], pr2 = b_w[2];
        #pragma unroll 8
        for (int r = 0; r < CRDIM; ++r) {
            float rv = rs[r * NP + p];
            pr0 = fmaf(w_w[r],          rv, pr0);
            pr1 = fmaf(w_w[CRDIM + r],  rv, pr1);
            pr2 = fmaf(w_w[2*CRDIM + r], rv, pr2);
        }
        float s0 = sigmoidf_(pr0);
        float s1 = sigmoidf_(pr1);
        float s2 = sigmoidf_(pr2);
        float tden = fmaxf(s0 + s1 + s2, 1e-6f);
        float n0 = s0 / tden, n1 = s1 / tden, n2 = s2 / tden;
        int gp = p0 + p;
        int i  = gp >> 8;
        int j  = gp & 255;
        // mask on scanned index; sAT is pre-transposed for the vertical scan
        sB [(size_t)b * PPLANE + gp]         = n1 + (i >= 1 ? n0 : 0.0f) + (i <= SDIM - 2 ? n2 : 0.0f);
        sAT[(size_t)b * PPLANE + j * SDIM + i] = n1 + (j >= 1 ? n0 : 0.0f) + (j <= SDIM - 2 ? n2 : 0.0f);
    }
}

// ---------------------------------------------------------------------------
// Kernel 2v: vertical scan.  thread = (b,c,j); all accesses coalesced.
// h_i[j] = h_{i-1}[j] * sAT[b,i,j] + v[b,c,i,j];  o_v = h * u.
// ---------------------------------------------------------------------------
__global__ __launch_bounds__(256) void k2_scan_v(
    const float* __restrict__ v_in, const float* __restrict__ u_in,
    const float* __restrict__ sAT, float* __restrict__ ov)
{
    const int b = blockIdx.x >> 7;
    const int c = blockIdx.x & 127;
    const int j = threadIdx.x;
    const size_t plane = (size_t)(b * CDIM + c) * PPLANE;
    const float* vp = v_in + plane;
    const float* up = u_in + plane;
    const float* sp = sAT + (size_t)b * PPLANE;
    float* op = ov + plane;

    float h = 0.0f;
    #pragma unroll 4
    for (int i = 0; i < SDIM; ++i) {
        int off = i * SDIM + j;
        h = fmaf(h, sp[off], vp[off]);
        op[off] = h * up[off];
    }
}

// ---------------------------------------------------------------------------
// Kernel 2h: horizontal scan (reference never un-transposes o_h).
// h_i[j] = h_{i-1}[j] * sB[b,i,j] + v[b,c,j,i];  o_h[b,c,i,j] = h * u[b,c,j,i].
// Each thread streams its own row of v/u via float4; s reads + o_h writes
// are coalesced across the wave.
// ---------------------------------------------------------------------------
__global__ __launch_bounds__(256) void k2_scan_h(
    const float* __restrict__ v_in, const float* __restrict__ u_in,
    const float* __restrict__ sB, float* __restrict__ oh)
{
    const int b = blockIdx.x >> 7;
    const int c = blockIdx.x & 127;
    const int j = threadIdx.x;
    const size_t plane = (size_t)(b * CDIM + c) * PPLANE;
    const float* vrow = v_in + plane + (size_t)j * SDIM;
    const float* urow = u_in + plane + (size_t)j * SDIM;
    const float* sp   = sB + (size_t)b * PPLANE;
    float* op = oh + plane;

    float h = 0.0f;
    for (int i = 0; i < SDIM; i += 4) {
        if (i + 16 < SDIM) {
            __builtin_prefetch(vrow + i + 16, 0, 1);   // global_prefetch_b8
            __builtin_prefetch(urow + i + 16, 0, 1);
        }
        float va[4], ua[4];
        *(float4*)va = *(const float4*)(vrow + i);
        *(float4*)ua = *(const float4*)(urow + i);
        #pragma unroll
        for (int t = 0; t < 4; ++t) {
            float s = sp[(i + t) * SDIM + j];
            h = fmaf(h, s, va[t]);
            op[(i + t) * SDIM + j] = h * ua[t];
        }
    }
}

// ---------------------------------------------------------------------------
// Kernel wf: fold W_merge[128x512] -> wcomb[128x256]
// (concat is [o_v,o_v,o_h,o_h], so pairs of weight blocks just add).
// ---------------------------------------------------------------------------
__global__ __launch_bounds__(256) void k_wfold(
    const float* __restrict__ w_merge, float* __restrict__ wcomb)
{
    int idx = blockIdx.x * 256 + threadIdx.x;   // < 32768
    int o = idx >> 8;
    int c = idx & 255;
    float v;
    if (c < 128) v = w_merge[o * 512 + c]       + w_merge[o * 512 + 128 + c];
    else         v = w_merge[o * 512 + 128 + c] + w_merge[o * 512 + 256 + c];
    wcomb[idx] = v;
}

// ---------------------------------------------------------------------------
// Kernel 3: merge GEMM  out[b,:,p] = wcomb[128x256] * [o_v; o_h] + b_merge.
// LDS-staged in two 32KB phases; 4 persistent f32 accumulators per wave.
// ---------------------------------------------------------------------------
__global__ __launch_bounds__(256) void k3_merge(
    const float* __restrict__ ov, const float* __restrict__ oh,
    const float* __restrict__ wcomb, const float* __restrict__ b_merge,
    float* __restrict__ out)
{
    __shared__ float ts[CDIM * NP];   // 32 KB

    const int tid  = threadIdx.x;
    const int lane = tid & 31;
    const int wv   = tid >> 5;
    const int hl   = lane >> 4;
    const int ln   = lane & 15;

    const int tile = blockIdx.x;
    const int b    = tile >> 10;
    const int p0   = (tile & 1023) * NP;

    v8f acc0 = {}, acc1 = {}, acc2 = {}, acc3 = {};

    for (int phase = 0; phase < 2; ++phase) {
        const float* src = (phase == 0) ? ov : oh;
        const float* sbp = src + (size_t)b * CDIM * PPLANE;
        __syncthreads();          // all waves done reading previous ts
        #pragma unroll
        for (int it = 0; it < 8; ++it) {
            int li = it * 256 + tid;
            int c  = li >> 4;
            int f  = (li & 15) << 2;
            *(float4*)(ts + c * NP + f) =
                *(const float4*)(sbp + (size_t)c * PPLANE + p0 + f);
        }
        __syncthreads();

        const int cb = phase * 128;
        #pragma unroll
        for (int q = 0; q < 4; ++q) {
            int t  = q * 8 + wv;
            int m0 = (t >> 2) * 16;
            int n0 = (t & 3) * 16;
            v8f a = (q == 0) ? acc0 : (q == 1) ? acc1 : (q == 2) ? acc2 : acc3;
            for (int ks = 0; ks < 32; ++ks) {   // K = 128 per phase
                int c4 = 4 * ks;
                int kb = c4 + 2 * hl;
                v2f av = *(const v2f*)(wcomb + (m0 + ln) * 256 + cb + kb);
                v2f bv;
                bv.x = ts[kb * NP + n0 + ln];
                bv.y = ts[(kb + 1) * NP + n0 + ln];
                a = wmma_f32(av, bv, a);
            }
            if (q == 0) acc0 = a; else if (q == 1) acc1 = a;
            else if (q == 2) acc2 = a; else acc3 = a;
        }
    }

    #pragma unroll
    for (int q = 0; q < 4; ++q) {
        int t  = q * 8 + wv;
        int m0 = (t >> 2) * 16;
        int n0 = (t & 3) * 16;
        v8f a = (q == 0) ? acc0 : (q == 1) ? acc1 : (q == 2) ? acc2 : acc3;
        int gp = p0 + n0 + ln;
        #pragma unroll
        for (int g = 0; g < 8; ++g) {
            int m = m0 + g + 8 * hl;
            out[(size_t)(b * CDIM + m) * PPLANE + gp] = a[g] + b_merge[m];
        }
    }
}

// ---------------------------------------------------------------------------
extern "C" void kernel_launch(void* const* d_in, const int* in_sizes, int n_in,
                              void* d_out, int out_size, void* d_ws, size_t ws_size,
                              hipStream_t stream) {
    const float* x        = (const float*)d_in[0];
    const float* w_reduce = (const float*)d_in[1];
    const float* b_reduce = (const float*)d_in[2];
    const float* w_u      = (const float*)d_in[3];
    const float* b_u      = (const float*)d_in[4];
    const float* w_lam    = (const float*)d_in[5];
    const float* b_lam    = (const float*)d_in[6];
    const float* w_w      = (const float*)d_in[7];
    const float* b_w      = (const float*)d_in[8];
    // d_in[9] (w_mw) and d_in[10] (b_mw) are dead code in the reference.
    const float* w_merge  = (const float*)d_in[11];
    const float* b_merge  = (const float*)d_in[12];
    float* out = (float*)d_out;

    const size_t BIG = (size_t)BDIM * CDIM * PPLANE;   // 16,777,216 floats
    char* ws = (char*)d_ws;
    float* u_buf  = (float*)(ws);
    float* v_buf  = (float*)(ws + 1 * BIG * sizeof(float));
    float* ov_buf = (float*)(ws + 2 * BIG * sizeof(float));
    float* oh_buf = (float*)(ws + 3 * BIG * sizeof(float));
    float* sAT    = (float*)(ws + 4 * BIG * sizeof(float));
    float* sB     = (float*)(ws + 4 * BIG * sizeof(float) + (size_t)BDIM * PPLANE * sizeof(float));
    float* wcomb  = (float*)(ws + 4 * BIG * sizeof(float) + (size_t)2 * BDIM * PPLANE * sizeof(float));

    k_wfold<<<128, 256, 0, stream>>>(w_merge, wcomb);

    k1_pointwise<<<BDIM * (PPLANE / NP), 256, 0, stream>>>(
        x, w_reduce, b_reduce, w_u, b_u, w_lam, b_lam, w_w, b_w,
        u_buf, v_buf, sAT, sB);

    k2_scan_v<<<BDIM * CDIM, 256, 0, stream>>>(v_buf, u_buf, sAT, ov_buf);
    k2_scan_h<<<BDIM * CDIM, 256, 0, stream>>>(v_buf, u_buf, sB, oh_buf);

    k3_merge<<<BDIM * (PPLANE / NP), 256, 0, stream>>>(
        ov_buf, oh_buf, wcomb, b_merge, out);
}